// Classifier_15453292331187
// MI455X (gfx1250) — compile-verified
//
#include <hip/hip_runtime.h>
#include <hip/hip_bf16.h>
#include <math.h>

typedef float v2f __attribute__((ext_vector_type(2)));
typedef float v4f __attribute__((ext_vector_type(4)));
typedef float v8f __attribute__((ext_vector_type(8)));

#define HDIM   128
#define NGRAPH 64

static __device__ __forceinline__ float warpSum32(float v) {
#pragma unroll
    for (int m = 16; m >= 1; m >>= 1) v += __shfl_xor(v, m, 32);
    return v;
}

// ---------------------------------------------------------------- zero
__global__ void k_zero(int* cnt, int* outdeg, float* poolG, float* cntG, int N) {
    int i = blockIdx.x * blockDim.x + threadIdx.x;
    int stride = gridDim.x * blockDim.x;
    for (int j = i; j < N; j += stride) { cnt[j] = 0; outdeg[j] = 0; }
    for (int j = i; j < NGRAPH * HDIM; j += stride) poolG[j] = 0.f;
    for (int j = i; j < NGRAPH; j += stride) cntG[j] = 0.f;
}

// ---------------------------------------------------------------- degree histograms
__global__ void k_count(const int* __restrict__ esrc, const int* __restrict__ edst,
                        int* cnt, int* outdeg, int E) {
    int i = blockIdx.x * blockDim.x + threadIdx.x;
    if (i < E) {
        atomicAdd(&cnt[edst[i]], 1);
        atomicAdd(&outdeg[esrc[i]], 1);
    }
}

// ---------------------------------------------------------------- single-block exclusive scan
__global__ void k_scan(const int* __restrict__ cnt, int* row_start, int* ptr, int N, int E) {
    __shared__ int tmp[1024];
    __shared__ int carry_s;
    int t = threadIdx.x;
    if (t == 0) carry_s = 0;
    __syncthreads();
    for (int base = 0; base < N; base += 1024) {
        int x = (base + t < N) ? cnt[base + t] : 0;
        tmp[t] = x;
        __syncthreads();
#pragma unroll
        for (int off = 1; off < 1024; off <<= 1) {
            int v = (t >= off) ? tmp[t - off] : 0;
            __syncthreads();
            tmp[t] += v;
            __syncthreads();
        }
        int excl = carry_s + tmp[t] - x;
        if (base + t < N) { row_start[base + t] = excl; ptr[base + t] = excl; }
        __syncthreads();
        if (t == 0) carry_s += tmp[1023];
        __syncthreads();
    }
    if (t == 0) row_start[N] = E;
}

// ---------------------------------------------------------------- CSR fill (by dst)
__global__ void k_fill(const int* __restrict__ esrc, const int* __restrict__ edst,
                       int* ptr, int* csr_src, int E) {
    int i = blockIdx.x * blockDim.x + threadIdx.x;
    if (i < E) {
        int p = atomicAdd(&ptr[edst[i]], 1);
        csr_src[p] = esrc[i];
    }
}

// ---------------------------------------------------------------- s = in_deg * out_deg^-0.5
__global__ void k_s(const int* __restrict__ row_start, const int* __restrict__ outdeg,
                    float* s, int N) {
    int n = blockIdx.x * blockDim.x + threadIdx.x;
    if (n < N) {
        float indeg = (float)(row_start[n + 1] - row_start[n]);
        float od = fmaxf((float)outdeg[n], 1.f);
        s[n] = indeg * rsqrtf(od);
    }
}

// ---------------------------------------------------------------- layer 1: wave per node
// h1s[n][f] = relu( (sum_in s[src]) * indeg^-0.5 * W1[f] + b1[f] ) * outdeg^-0.5
__global__ __launch_bounds__(256) void k_layer1(
    const int* __restrict__ row_start, const int* __restrict__ csr_src,
    const int* __restrict__ outdeg, const float* __restrict__ s,
    const float* __restrict__ W1, const float* __restrict__ b1,
    float* __restrict__ h1s, int N) {
    int lane = threadIdx.x & 31;
    int wave = threadIdx.x >> 5;
    int n = blockIdx.x * 8 + wave;
    if (n >= N) return;
    int beg = row_start[n], end = row_start[n + 1];
    float sum = 0.f;
    for (int i = beg + lane; i < end; i += 32) sum += s[csr_src[i]];
    sum = warpSum32(sum);
    float a    = sum * rsqrtf(fmaxf((float)(end - beg), 1.f));
    float nsrc = rsqrtf(fmaxf((float)outdeg[n], 1.f));
    v4f w  = *(const v4f*)(W1 + lane * 4);
    v4f bb = *(const v4f*)(b1 + lane * 4);
    v4f o;
    o.x = fmaxf(fmaf(a, w.x, bb.x), 0.f) * nsrc;
    o.y = fmaxf(fmaf(a, w.y, bb.y), 0.f) * nsrc;
    o.z = fmaxf(fmaf(a, w.z, bb.z), 0.f) * nsrc;
    o.w = fmaxf(fmaf(a, w.w, bb.w), 0.f) * nsrc;
    *(v4f*)(h1s + (size_t)n * HDIM + lane * 4) = o;
}

// ---------------------------------------------------------------- fused layer-2 aggregation + WMMA GEMM
// Block = 256 threads (8 waves), handles 16 nodes (rows). Phase 1: each wave
// CSR-aggregates 2 rows of 128-dim features into LDS (scaled by norm_dst).
// Phase 2: wave w computes the 16x16 output tile at cols [16w,16w+16) via
// 32 x V_WMMA_F32_16X16X4_F32 over K=128, C preloaded with bias, ReLU on store.
#define AS_PITCH 132
__global__ __launch_bounds__(256) void k_gemm(
    const int* __restrict__ row_start, const int* __restrict__ csr_src,
    const float* __restrict__ h1s, const float* __restrict__ W2,
    const float* __restrict__ b2, float* __restrict__ h2, int N) {
    __shared__ float As[16][AS_PITCH];
    int lane = threadIdx.x & 31;
    int wave = threadIdx.x >> 5;
    int r0 = blockIdx.x * 16;

    // phase 1: aggregation into LDS
#pragma unroll
    for (int rr = 0; rr < 2; ++rr) {
        int row = wave * 2 + rr;
        int n = r0 + row;
        v4f acc = {0.f, 0.f, 0.f, 0.f};
        if (n < N) {
            int beg = row_start[n], end = row_start[n + 1];
            for (int i = beg; i < end; ++i) {
                int src = csr_src[i];  // uniform across wave -> scalar broadcast
                v4f v = *(const v4f*)(h1s + (size_t)src * HDIM + lane * 4);
                acc.x += v.x; acc.y += v.y; acc.z += v.z; acc.w += v.w;
            }
            float nd = rsqrtf(fmaxf((float)(end - beg), 1.f));
            acc.x *= nd; acc.y *= nd; acc.z *= nd; acc.w *= nd;
        }
        *(v4f*)(&As[row][lane * 4]) = acc;
    }
    __syncthreads();

    // phase 2: WMMA 16x16 tile, K = 128 in steps of 4
    int c0 = wave * 16;
    int half = lane >> 4;   // 0: K{0,1}, 1: K{2,3} per 32-bit A/B layout
    int l = lane & 15;
    float bias = b2[c0 + l];
    v8f acc;
#pragma unroll
    for (int v = 0; v < 8; ++v) acc[v] = bias;

#pragma unroll
    for (int k0 = 0; k0 < HDIM; k0 += 4) {
        int ka = k0 + 2 * half;
        v2f a, b;
        a.x = As[l][ka];
        a.y = As[l][ka + 1];
        b.x = W2[(size_t)ka * HDIM + c0 + l];
        b.y = W2[(size_t)(ka + 1) * HDIM + c0 + l];
        acc = __builtin_amdgcn_wmma_f32_16x16x4_f32(
            /*neg_a=*/false, a, /*neg_b=*/false, b,
            /*c_mod=*/(short)0, acc, /*reuse_a=*/false, /*reuse_b=*/false);
    }

#pragma unroll
    for (int v = 0; v < 8; ++v) {
        int n = r0 + v + 8 * half;   // M = v + 8*(lane>=16)
        if (n < N) h2[(size_t)n * HDIM + c0 + l] = fmaxf(acc[v], 0.f);
    }
}

// ---------------------------------------------------------------- graph mean-pool (LDS accumulate)
__global__ __launch_bounds__(256) void k_pool(
    const float* __restrict__ h2, const int* __restrict__ gids,
    float* poolG, float* cntG, int N) {
    __shared__ float poolL[NGRAPH * HDIM];
    __shared__ float cntL[NGRAPH];
    int t = threadIdx.x;
    for (int j = t; j < NGRAPH * HDIM; j += blockDim.x) poolL[j] = 0.f;
    for (int j = t; j < NGRAPH; j += blockDim.x) cntL[j] = 0.f;
    __syncthreads();
    int lane = t & 31, wave = t >> 5;
    int stride = gridDim.x * 8;
    for (int n = blockIdx.x * 8 + wave; n < N; n += stride) {
        int g = gids[n];
        v4f v = *(const v4f*)(h2 + (size_t)n * HDIM + lane * 4);
        atomicAdd(&poolL[g * HDIM + lane * 4 + 0], v.x);
        atomicAdd(&poolL[g * HDIM + lane * 4 + 1], v.y);
        atomicAdd(&poolL[g * HDIM + lane * 4 + 2], v.z);
        atomicAdd(&poolL[g * HDIM + lane * 4 + 3], v.w);
        if (lane == 0) atomicAdd(&cntL[g], 1.f);
    }
    __syncthreads();
    for (int j = t; j < NGRAPH * HDIM; j += blockDim.x)
        if (poolL[j] != 0.f) atomicAdd(&poolG[j], poolL[j]);
    for (int j = t; j < NGRAPH; j += blockDim.x)
        if (cntL[j] != 0.f) atomicAdd(&cntG[j], cntL[j]);
}

// ---------------------------------------------------------------- readout: wave per graph
__global__ __launch_bounds__(256) void k_final(
    const float* __restrict__ poolG, const float* __restrict__ cntG,
    const float* __restrict__ Wl, const float* __restrict__ bl,
    float* __restrict__ out) {
    int lane = threadIdx.x & 31;
    int wave = threadIdx.x >> 5;
    int g = blockIdx.x * 8 + wave;
    if (g >= NGRAPH) return;
    float c = fmaxf(cntG[g], 1.f);
    float p = 0.f;
#pragma unroll
    for (int j = 0; j < 4; ++j) {
        int f = lane * 4 + j;
        p += (poolG[g * HDIM + f] / c) * Wl[f];
    }
    p = warpSum32(p);
    if (lane == 0) out[g] = 1.f / (1.f + expf(-(p + bl[0])));
}

// ---------------------------------------------------------------- launch
static inline size_t alignUp(size_t x) { return (x + 255) & ~(size_t)255; }

extern "C" void kernel_launch(void* const* d_in, const int* in_sizes, int n_in,
                              void* d_out, int out_size, void* d_ws, size_t ws_size,
                              hipStream_t stream) {
    const int* esrc = (const int*)d_in[0];
    const int* edst = (const int*)d_in[1];
    const int* gids = (const int*)d_in[2];
    const float* W1 = (const float*)d_in[3];
    const float* b1 = (const float*)d_in[4];
    const float* W2 = (const float*)d_in[5];
    const float* b2 = (const float*)d_in[6];
    const float* Wl = (const float*)d_in[7];
    const float* bl = (const float*)d_in[8];
    float* out = (float*)d_out;

    const int E = in_sizes[0];
    const int N = in_sizes[2];

    // workspace layout
    char* ws = (char*)d_ws;
    size_t off = 0;
    int* cnt       = (int*)(ws + off); off = alignUp(off + (size_t)N * 4);
    int* outdeg    = (int*)(ws + off); off = alignUp(off + (size_t)N * 4);
    int* row_start = (int*)(ws + off); off = alignUp(off + ((size_t)N + 1) * 4);
    int* ptr       = (int*)(ws + off); off = alignUp(off + (size_t)N * 4);
    int* csr_src   = (int*)(ws + off); off = alignUp(off + (size_t)E * 4);
    float* s       = (float*)(ws + off); off = alignUp(off + (size_t)N * 4);
    float* h1s     = (float*)(ws + off); off = alignUp(off + (size_t)N * HDIM * 4);
    float* h2      = (float*)(ws + off); off = alignUp(off + (size_t)N * HDIM * 4);
    float* poolG   = (float*)(ws + off); off = alignUp(off + (size_t)NGRAPH * HDIM * 4);
    float* cntG    = (float*)(ws + off); off = alignUp(off + (size_t)NGRAPH * 4);
    (void)ws_size;

    const int T = 256;
    int gE = (E + T - 1) / T;
    int gN = (N + T - 1) / T;
    int gW = (N + 7) / 8;        // one wave per node
    int gR = (N + 15) / 16;      // 16 rows per GEMM block

    k_zero  <<<512, T, 0, stream>>>(cnt, outdeg, poolG, cntG, N);
    k_count <<<gE, T, 0, stream>>>(esrc, edst, cnt, outdeg, E);
    k_scan  <<<1, 1024, 0, stream>>>(cnt, row_start, ptr, N, E);
    k_fill  <<<gE, T, 0, stream>>>(esrc, edst, ptr, csr_src, E);
    k_s     <<<gN, T, 0, stream>>>(row_start, outdeg, s, N);
    k_layer1<<<gW, T, 0, stream>>>(row_start, csr_src, outdeg, s, W1, b1, h1s, N);
    k_gemm  <<<gR, T, 0, stream>>>(row_start, csr_src, h1s, W2, b2, h2, N);
    k_pool  <<<64, T, 0, stream>>>(h2, gids, poolG, cntG, N);
    k_final <<<(NGRAPH + 7) / 8, T, 0, stream>>>(poolG, cntG, Wl, bl, out);
    (void)out_size; (void)n_in;
}